// EVSSM_9234179686991
// MI455X (gfx1250) — compile-verified
//
#include <hip/hip_runtime.h>
#include <math.h>

#define B_    2
#define CIN   64
#define H_    128
#define W_    128
#define L_    (H_*W_)        // 16384
#define DIN   128            // D_INNER
#define NST   8              // D_STATE
#define DTR   4
#define CPROJ 20             // DT_RANK + 2*D_STATE
#define HID   192
#define HID2  384
#define NPIX  (B_*L_)        // 32768

typedef __attribute__((ext_vector_type(16))) _Float16 v16h;
typedef __attribute__((ext_vector_type(8)))  float    v8f;
typedef __attribute__((ext_vector_type(2)))  float    v2f;

__device__ __forceinline__ float gelu_f(float x) {
    return 0.5f * x * (1.0f + erff(x * 0.70710678118654752f));
}
__device__ __forceinline__ float softplus_f(float x) {
    return (x > 20.f) ? x : log1pf(expf(x));
}

// ---------------------------------------------------------------------------
// WMMA GEMM: out[M x N] = A[M x K] * W[N x K]^T, exact fp32 via
// V_WMMA_F32_16X16X4_F32.  Each wave computes an MT x NT block of 16x16
// tiles; fragment loads are double-buffered so step s+1's loads overlap
// step s's WMMAs.  K compile-time -> fully unrolled.
// MODE 0: out row-major (M x N). MODE 1: out NCHW (B_, N, H_, W_) += res.
// ---------------------------------------------------------------------------
template<int K, int MT, int NT, int MODE>
__global__ __launch_bounds__(32) void gemm_wmma_kernel(
    const float* __restrict__ A, const float* __restrict__ Wt,
    float* __restrict__ out, int N, const float* __restrict__ res)
{
    const int lane = threadIdx.x;
    const int lo   = lane & 15;
    const int hi   = lane >> 4;
    const int mt0  = blockIdx.x * MT;     // first m-tile
    const int ng   = blockIdx.y;          // group of NT n-tiles

    const float* __restrict__ arow[MT];
#pragma unroll
    for (int m = 0; m < MT; ++m)
        arow[m] = A + (size_t)((mt0 + m) * 16 + lo) * K + 2 * hi;
    const float* __restrict__ brow[NT];
#pragma unroll
    for (int t = 0; t < NT; ++t)
        brow[t] = Wt + (size_t)((ng * NT + t) * 16 + lo) * K + 2 * hi;

    v8f acc[MT][NT];
#pragma unroll
    for (int m = 0; m < MT; ++m)
#pragma unroll
        for (int t = 0; t < NT; ++t) acc[m][t] = (v8f){};

    constexpr int STEPS = K / 4;
    v2f af[2][MT], bf[2][NT];

    // prologue: fragments for step 0
#pragma unroll
    for (int m = 0; m < MT; ++m) af[0][m] = *(const v2f*)arow[m];
#pragma unroll
    for (int t = 0; t < NT; ++t) bf[0][t] = *(const v2f*)brow[t];

#pragma unroll
    for (int s = 0; s < STEPS; ++s) {
        const int cur = s & 1, nxt = cur ^ 1;
        if (s + 1 < STEPS) {
            const int kt = (s + 1) * 4;
#pragma unroll
            for (int m = 0; m < MT; ++m) af[nxt][m] = *(const v2f*)(arow[m] + kt);
#pragma unroll
            for (int t = 0; t < NT; ++t) bf[nxt][t] = *(const v2f*)(brow[t] + kt);
        }
#pragma unroll
        for (int m = 0; m < MT; ++m)
#pragma unroll
            for (int t = 0; t < NT; ++t)
                acc[m][t] = __builtin_amdgcn_wmma_f32_16x16x4_f32(
                    false, af[cur][m], false, bf[cur][t],
                    (short)0, acc[m][t], false, false);
    }

#pragma unroll
    for (int m = 0; m < MT; ++m) {
#pragma unroll
        for (int t = 0; t < NT; ++t) {
            const int ncol = (ng * NT + t) * 16 + lo;
#pragma unroll
            for (int r = 0; r < 8; ++r) {
                int mrow = (mt0 + m) * 16 + r + hi * 8;
                float v  = acc[m][t][r];
                if (MODE == 0) {
                    out[(size_t)mrow * N + ncol] = v;
                } else {
                    int b = mrow >> 14;
                    int l = mrow & (L_ - 1);
                    size_t o = ((size_t)(b * N + ncol) << 14) + l;
                    out[o] = v + res[o];
                }
            }
        }
    }
}

// ---------------------------------------------------------------------------
// Flip(H,W) + channel RMS-norm (wb_ln). Writes flipped NCHW copy and the
// normalized pixel-major (NPIX x 64) activation.
// ---------------------------------------------------------------------------
__global__ __launch_bounds__(128) void flip_rmsnorm_kernel(
    const float* __restrict__ x, const float* __restrict__ w,
    const float* __restrict__ b, float* __restrict__ xflip,
    float* __restrict__ tn)
{
    int p = blockIdx.x * blockDim.x + threadIdx.x;
    if (p >= NPIX) return;
    int bb = p >> 14, l = p & (L_ - 1);
    int h = l >> 7, wq = l & 127;
    int hs = (H_ - 1) - h, ws = (W_ - 1) - wq;
    const float* src = x + (((size_t)bb * CIN) << 14) + (hs << 7) + ws;
    float v[CIN];
    float ss = 0.f;
#pragma unroll
    for (int c = 0; c < CIN; ++c) { float t = src[(size_t)c << 14]; v[c] = t; ss += t * t; }
    float sc = rsqrtf(ss * (1.0f / CIN) + 1e-6f);
    float* xf = xflip + (((size_t)bb * CIN) << 14) + l;
#pragma unroll
    for (int c = 0; c < CIN; ++c) {
        xf[(size_t)c << 14]     = v[c];
        tn[(size_t)p * CIN + c] = v[c] * sc * w[c] + b[c];
    }
}

// Channel RMS-norm (no flip): NCHW in -> pixel-major (NPIX x 64) out.
__global__ __launch_bounds__(128) void rmsnorm_kernel(
    const float* __restrict__ in, const float* __restrict__ w,
    const float* __restrict__ b, float* __restrict__ outp)
{
    int p = blockIdx.x * blockDim.x + threadIdx.x;
    if (p >= NPIX) return;
    int bb = p >> 14, l = p & (L_ - 1);
    const float* src = in + (((size_t)bb * CIN) << 14) + l;
    float ss = 0.f;
#pragma unroll
    for (int c = 0; c < CIN; ++c) { float t = src[(size_t)c << 14]; ss += t * t; }
    float sc = rsqrtf(ss * (1.0f / CIN) + 1e-6f);
#pragma unroll
    for (int c = 0; c < CIN; ++c)
        outp[(size_t)p * CIN + c] = src[(size_t)c << 14] * sc * w[c] + b[c];
}

// ---------------------------------------------------------------------------
// Depthwise 3x3 conv (pad 1) on xin (first 128 cols of pixel-major xz) + bias
// + exact gelu.  Output xs in (b, d, l) channel-major layout.
// ---------------------------------------------------------------------------
__global__ void conv2d_gelu_kernel(const float* __restrict__ xz,
                                   const float* __restrict__ cw,
                                   const float* __restrict__ cb,
                                   float* __restrict__ xs)
{
    int idx = blockIdx.x * blockDim.x + threadIdx.x;
    if (idx >= B_ * DIN * L_) return;
    int l  = idx & (L_ - 1);
    int d  = (idx >> 14) & (DIN - 1);
    int bb = idx >> 21;
    int h = l >> 7, w = l & 127;
    float s = cb[d];
#pragma unroll
    for (int kh = 0; kh < 3; ++kh) {
        int hh = h + kh - 1; if (hh < 0 || hh >= H_) continue;
#pragma unroll
        for (int kw = 0; kw < 3; ++kw) {
            int ww = w + kw - 1; if (ww < 0 || ww >= W_) continue;
            int pp = (bb << 14) + (hh << 7) + ww;
            s += xz[(size_t)pp * 256 + d] * cw[d * 9 + kh * 3 + kw];
        }
    }
    xs[idx] = gelu_f(s);
}

// x_proj: xdbl[b,c,l] = sum_d W[c,d] * xs[b,d,l]
__global__ void xproj_kernel(const float* __restrict__ xs,
                             const float* __restrict__ W,
                             float* __restrict__ xdbl)
{
    int idx = blockIdx.x * blockDim.x + threadIdx.x;
    if (idx >= B_ * CPROJ * L_) return;
    int l  = idx & (L_ - 1);
    int c  = (idx >> 14) % CPROJ;
    int bb = idx / (CPROJ << 14);
    const float* xp = xs + (((size_t)bb * DIN) << 14) + l;
    float s = 0.f;
#pragma unroll 8
    for (int d = 0; d < DIN; ++d) s += W[c * DIN + d] * xp[(size_t)d << 14];
    xdbl[idx] = s;
}

// Depthwise 1D conv k=7 pad=3 + bias on (B, 20, L)
__global__ void conv1d_kernel(const float* __restrict__ xdbl,
                              const float* __restrict__ cw,
                              const float* __restrict__ cb,
                              float* __restrict__ xdbl2)
{
    int idx = blockIdx.x * blockDim.x + threadIdx.x;
    if (idx >= B_ * CPROJ * L_) return;
    int l = idx & (L_ - 1);
    int c = (idx >> 14) % CPROJ;
    float s = cb[c];
    const float* row = xdbl + (size_t)(idx - l);
#pragma unroll
    for (int k = 0; k < 7; ++k) {
        int l2 = l + k - 3;
        if (l2 >= 0 && l2 < L_) s += row[l2] * cw[c * 7 + k];
    }
    xdbl2[idx] = s;
}

// dts[b,d,l] = sum_r xdbl2[b,r,l] * dtw[d,r]   (r < 4)
__global__ void dtproj_kernel(const float* __restrict__ xdbl2,
                              const float* __restrict__ dtw,
                              float* __restrict__ dts)
{
    int idx = blockIdx.x * blockDim.x + threadIdx.x;
    if (idx >= B_ * DIN * L_) return;
    int l  = idx & (L_ - 1);
    int d  = (idx >> 14) & (DIN - 1);
    int bb = idx >> 21;
    size_t base = ((size_t)(bb * CPROJ) << 14) + l;
    float s = 0.f;
#pragma unroll
    for (int r = 0; r < DTR; ++r) s += xdbl2[base + ((size_t)r << 14)] * dtw[d * DTR + r];
    dts[idx] = s;
}

// ---------------------------------------------------------------------------
// Selective scan: one lane per (b,d) chain, 8 recurrent states in registers.
// Latency-bound tail (only 256 chains) -> prefetch streaming lines ahead.
// ---------------------------------------------------------------------------
__global__ __launch_bounds__(32) void scan_kernel(
    const float* __restrict__ xs, const float* __restrict__ dts,
    const float* __restrict__ xdbl2, const float* __restrict__ A_logs,
    const float* __restrict__ Ds, const float* __restrict__ dt_bias,
    float* __restrict__ yscan)
{
    int t = blockIdx.x * 32 + threadIdx.x;
    if (t >= B_ * DIN) return;
    int bb = t >> 7, d = t & (DIN - 1);
    float A[NST], h[NST];
#pragma unroll
    for (int n = 0; n < NST; ++n) { A[n] = -expf(A_logs[d * NST + n]); h[n] = 0.f; }
    float db = dt_bias[d], Dd = Ds[d];
    size_t xbase = ((size_t)(bb * DIN + d)) << 14;
    size_t Bbase = ((size_t)(bb * CPROJ + DTR)) << 14;
    size_t Cbase = ((size_t)(bb * CPROJ + DTR + NST)) << 14;
    for (int l = 0; l < L_; ++l) {
        if ((l & 127) == 0 && l + 512 < L_) {
            __builtin_prefetch(&dts[xbase + l + 512], 0, 1);
            __builtin_prefetch(&xs[xbase + l + 512], 0, 1);
            __builtin_prefetch(&xdbl2[Bbase + l + 512], 0, 1);
            __builtin_prefetch(&xdbl2[Cbase + l + 512], 0, 1);
        }
        float dtv = softplus_f(dts[xbase + l] + db);
        float xv  = xs[xbase + l];
        float dx  = dtv * xv;
        float y   = 0.f;
#pragma unroll
        for (int n = 0; n < NST; ++n) {
            float bn = xdbl2[Bbase + ((size_t)n << 14) + l];
            float cn = xdbl2[Cbase + ((size_t)n << 14) + l];
            h[n] = expf(dtv * A[n]) * h[n] + dx * bn;
            y   += h[n] * cn;
        }
        yscan[xbase + l] = y + Dd * xv;
    }
}

// LayerNorm over 128 channels x gelu(z) -> pixel-major (NPIX x 128)
__global__ __launch_bounds__(128) void lnz_kernel(
    const float* __restrict__ yscan, const float* __restrict__ xz,
    const float* __restrict__ ow, const float* __restrict__ ob,
    float* __restrict__ ty)
{
    int p = blockIdx.x * blockDim.x + threadIdx.x;
    if (p >= NPIX) return;
    int bb = p >> 14, l = p & (L_ - 1);
    const float* yp = yscan + (((size_t)bb * DIN) << 14) + l;
    float s = 0.f, s2 = 0.f;
#pragma unroll 8
    for (int d = 0; d < DIN; ++d) {
        float v = yp[(size_t)d << 14];
        s += v; s2 += v * v;
    }
    float mu  = s * (1.0f / DIN);
    float var = s2 * (1.0f / DIN) - mu * mu;
    float rs  = rsqrtf(var + 1e-5f);
    const float* zp = xz + (size_t)p * 256 + DIN;
#pragma unroll 8
    for (int d = 0; d < DIN; ++d) {
        float v = (yp[(size_t)d << 14] - mu) * rs * ow[d] + ob[d];
        ty[(size_t)p * DIN + d] = v * gelu_f(zp[d]);
    }
}

// Depthwise 3x3 (no bias) on h2 (pixel-major 384ch) + GLU: gelu(x1)*x2
__global__ void conv2_glu_kernel(const float* __restrict__ h2,
                                 const float* __restrict__ dw,
                                 float* __restrict__ t3)
{
    int idx = blockIdx.x * blockDim.x + threadIdx.x;
    if (idx >= NPIX * HID) return;
    int c = idx % HID;
    int p = idx / HID;
    int bb = p >> 14, l = p & (L_ - 1);
    int h = l >> 7, w = l & 127;
    float a = 0.f, g2 = 0.f;
#pragma unroll
    for (int kh = 0; kh < 3; ++kh) {
        int hh = h + kh - 1; if (hh < 0 || hh >= H_) continue;
#pragma unroll
        for (int kw = 0; kw < 3; ++kw) {
            int ww = w + kw - 1; if (ww < 0 || ww >= W_) continue;
            int pp = (bb << 14) + (hh << 7) + ww;
            a  += h2[(size_t)pp * HID2 + c]       * dw[c * 9 + kh * 3 + kw];
            g2 += h2[(size_t)pp * HID2 + HID + c] * dw[(HID + c) * 9 + kh * 3 + kw];
        }
    }
    t3[idx] = gelu_f(a) * g2;
}

// ---------------------------------------------------------------------------
// Per-8x8-patch real FFT filter: irfft2(rfft2(patch) * P) + residual.
// One thread per (b, ch, patch); direct 8-point DFT with exact twiddle table.
// ---------------------------------------------------------------------------
__global__ __launch_bounds__(32) void fftpatch_kernel(
    const float* __restrict__ y2, const float* __restrict__ fp,
    const float* __restrict__ res1, float* __restrict__ out)
{
    int t = blockIdx.x * 32 + threadIdx.x;
    if (t >= B_ * CIN * 16 * 16) return;
    int pw = t & 15;
    int ph = (t >> 4) & 15;
    int ch = (t >> 8) & (CIN - 1);
    int bb = t >> 14;

    const float r2 = 0.70710678118654752f;
    const float ct[8] = {1.f,  r2, 0.f, -r2, -1.f, -r2, 0.f,  r2};
    const float st[8] = {0.f,  r2, 1.f,  r2,  0.f, -r2, -1.f, -r2};

    float pr[8][8];
    for (int yy = 0; yy < 8; ++yy)
        for (int xx = 0; xx < 8; ++xx) {
            int h = ph * 8 + yy, w = pw * 8 + xx;
            pr[yy][xx] = y2[(size_t)((bb << 14) + (h << 7) + w) * CIN + ch];
        }
    float Fr[8][8], Fi[8][8];
    for (int yy = 0; yy < 8; ++yy)
        for (int v = 0; v < 8; ++v) {
            float re = 0.f, im = 0.f;
            for (int xx = 0; xx < 8; ++xx) {
                int k = (v * xx) & 7;
                re += pr[yy][xx] * ct[k];
                im -= pr[yy][xx] * st[k];
            }
            Fr[yy][v] = re; Fi[yy][v] = im;
        }
    float Gr[8][8], Gi[8][8];
    for (int u = 0; u < 8; ++u)
        for (int v = 0; v < 8; ++v) {
            float re = 0.f, im = 0.f;
            for (int yy = 0; yy < 8; ++yy) {
                int k = (u * yy) & 7;
                re += Fr[yy][v] * ct[k] + Fi[yy][v] * st[k];
                im += Fi[yy][v] * ct[k] - Fr[yy][v] * st[k];
            }
            float pv = (v <= 4) ? fp[ch * 40 + u * 5 + v]
                                : fp[ch * 40 + (((8 - u) & 7) * 5) + (8 - v)];
            Gr[u][v] = re * pv; Gi[u][v] = im * pv;
        }
    for (int yy = 0; yy < 8; ++yy)
        for (int xx = 0; xx < 8; ++xx) {
            float acc = 0.f;
            for (int u = 0; u < 8; ++u)
                for (int v = 0; v < 8; ++v) {
                    int k = (u * yy + v * xx) & 7;
                    acc += Gr[u][v] * ct[k] - Gi[u][v] * st[k];
                }
            size_t o = (((size_t)(bb * CIN + ch)) << 14)
                     + ((ph * 8 + yy) << 7) + (pw * 8 + xx);
            out[o] = res1[o] + acc * (1.0f / 64.0f);
        }
}

// ---------------------------------------------------------------------------
extern "C" void kernel_launch(void* const* d_in, const int* in_sizes, int n_in,
                              void* d_out, int out_size, void* d_ws, size_t ws_size,
                              hipStream_t stream)
{
    const float* x          = (const float*)d_in[0];
    const float* norm1_w    = (const float*)d_in[1];
    const float* norm1_b    = (const float*)d_in[2];
    const float* in_proj_w  = (const float*)d_in[3];
    const float* conv2d_w   = (const float*)d_in[4];
    const float* conv2d_b   = (const float*)d_in[5];
    const float* x_proj_w   = (const float*)d_in[6];
    const float* x_conv_w   = (const float*)d_in[7];
    const float* x_conv_b   = (const float*)d_in[8];
    const float* dt_projs_w = (const float*)d_in[9];
    const float* dt_projs_b = (const float*)d_in[10];
    const float* A_logs     = (const float*)d_in[11];
    const float* Ds         = (const float*)d_in[12];
    const float* out_norm_w = (const float*)d_in[13];
    const float* out_norm_b = (const float*)d_in[14];
    const float* out_proj_w = (const float*)d_in[15];
    const float* norm2_w    = (const float*)d_in[16];
    const float* norm2_b    = (const float*)d_in[17];
    const float* pin_w      = (const float*)d_in[18];
    const float* dw_w       = (const float*)d_in[19];
    const float* fft_p      = (const float*)d_in[20];
    const float* pout_w     = (const float*)d_in[21];
    float* out = (float*)d_out;

    // workspace layout (floats), with reuse
    float* base  = (float*)d_ws;
    float* segA  = base;                           // xflip (later y2)
    float* segB  = segA + (size_t)NPIX * CIN;      // t_nrm (later t2)
    float* segC  = segB + (size_t)NPIX * CIN;      // xz (later t3)
    float* segD  = segC + (size_t)NPIX * 256;      // scan block (later h2)
    float* xs    = segD;
    float* xdbl  = xs    + (size_t)B_ * DIN * L_;
    float* xdbl2 = xdbl  + (size_t)B_ * CPROJ * L_;
    float* dts   = xdbl2 + (size_t)B_ * CPROJ * L_;
    float* yscan = dts   + (size_t)B_ * DIN * L_;
    float* ty    = yscan + (size_t)B_ * DIN * L_;
    float* segE  = ty    + (size_t)NPIX * DIN;     // res1
    float* xflip = segA, *y2 = segA;
    float* tnrm  = segB, *t2 = segB;
    float* xz    = segC, *t3 = segC;
    float* h2    = segD;
    float* res1  = segE;

    // 1) flip + wb_ln(norm1)
    flip_rmsnorm_kernel<<<(NPIX + 127) / 128, 128, 0, stream>>>(x, norm1_w, norm1_b, xflip, tnrm);
    // 2) in_proj: (NPIX x 256)  K=64; MT=2 m-tiles, NT=4 n-tiles per wave
    gemm_wmma_kernel<64, 2, 4, 0><<<dim3(NPIX / 32, 4), 32, 0, stream>>>(tnrm, in_proj_w, xz, 256, nullptr);
    // 3) depthwise 3x3 + gelu -> xs (b,d,l)
    conv2d_gelu_kernel<<<(B_ * DIN * L_) / 256, 256, 0, stream>>>(xz, conv2d_w, conv2d_b, xs);
    // 4) x_proj
    xproj_kernel<<<(B_ * CPROJ * L_) / 256, 256, 0, stream>>>(xs, x_proj_w, xdbl);
    // 5) depthwise 1D k=7
    conv1d_kernel<<<(B_ * CPROJ * L_) / 256, 256, 0, stream>>>(xdbl, x_conv_w, x_conv_b, xdbl2);
    // 6) dt projection
    dtproj_kernel<<<(B_ * DIN * L_) / 256, 256, 0, stream>>>(xdbl2, dt_projs_w, dts);
    // 7) selective scan (256 chains, 8 waves)
    scan_kernel<<<(B_ * DIN) / 32, 32, 0, stream>>>(xs, dts, xdbl2, A_logs, Ds, dt_projs_b, yscan);
    // 8) LN x gelu(z) -> ty (NPIX x 128)
    lnz_kernel<<<(NPIX + 127) / 128, 128, 0, stream>>>(yscan, xz, out_norm_w, out_norm_b, ty);
    // 9) out_proj + residual -> res1 (NCHW)   K=128
    gemm_wmma_kernel<128, 2, 4, 1><<<dim3(NPIX / 32, 1), 32, 0, stream>>>(ty, out_proj_w, res1, 64, xflip);
    // 10) wb_ln(norm2)
    rmsnorm_kernel<<<(NPIX + 127) / 128, 128, 0, stream>>>(res1, norm2_w, norm2_b, t2);
    // 11) pin: (NPIX x 384)  K=64
    gemm_wmma_kernel<64, 2, 4, 0><<<dim3(NPIX / 32, 6), 32, 0, stream>>>(t2, pin_w, h2, 384, nullptr);
    // 12) depthwise 3x3 + GLU -> t3 (NPIX x 192)
    conv2_glu_kernel<<<((size_t)NPIX * HID) / 256, 256, 0, stream>>>(h2, dw_w, t3);
    // 13) pout: (NPIX x 64)  K=192
    gemm_wmma_kernel<192, 2, 4, 0><<<dim3(NPIX / 32, 1), 32, 0, stream>>>(t3, pout_w, y2, 64, nullptr);
    // 14) per-patch FFT filter + residual -> d_out (NCHW)
    fftpatch_kernel<<<(B_ * CIN * 256) / 32, 32, 0, stream>>>(y2, fft_p, res1, out);
}